// Wavelet_81681688035616
// MI455X (gfx1250) — compile-verified
//
#include <hip/hip_runtime.h>
#include <cstddef>

typedef float v2f __attribute__((ext_vector_type(2)));
typedef float v8f __attribute__((ext_vector_type(8)));

#define NN    1024          // image side
#define HALF  512           // quadrant side
#define RTILE 16            // quadrant rows per block
#define CTILE 64            // quadrant cols per block
#define RI    44            // staged intermediate rows (2*RTILE + halo slack)
#define XSTR  140           // staged x cols / row stride (2*CTILE + halo slack)
#define CSTR  72            // cA/cD LDS row stride (2*72 mod 64 = 16 -> conflict-free halves)
#define KCH   11            // WMMA K-chunks of 4 -> K = 44 = RI

__global__ __launch_bounds__(256) void wav2d_kernel(
    const float* __restrict__ x, const float* __restrict__ bmt,
    const int* __restrict__ m1p, const int* __restrict__ m2p,
    float* __restrict__ out)
{
  __shared__ float xt[RI * XSTR];
  __shared__ float cAt[RI * CSTR];
  __shared__ float cDt[RI * CSTR];
  __shared__ float decLo[8];
  __shared__ float decHi[8];

  const int tid  = threadIdx.x;
  const int lane = tid & 31;
  const int wid  = tid >> 5;
  const int b    = blockIdx.z;
  const int r0   = blockIdx.y * RTILE;   // quadrant row base
  const int c0   = blockIdx.x * CTILE;   // quadrant col base

  const int m1    = m1p[0];
  const int m2    = m2p[0];
  const int offLo = m1;        // low-pass:  x[(2o + t - m1)   mod N]
  const int offHi = m2 - 1;    // high-pass: x[(2o + t - (m2-1)) mod N]
  const int pA    = (offLo > offHi) ? offLo : offHi;
  const int sLo   = pA - offLo;
  const int sHi   = pA - offHi;

  // dec_lo = flip(bmt); dec_hi[t] = (flip(bmt)*(-1)^j) flipped = bmt[t]*sign(7-t)
  if (tid < 8) {
    decLo[tid] = bmt[7 - tid];
    decHi[tid] = bmt[tid] * (((7 - tid) & 1) ? -1.0f : 1.0f);
  }

  // ---- stage 0: stage x tile into LDS (periodic wrap) ----
  const int rowBase = 2 * r0 - pA;
  const int colBase = 2 * c0 - pA;
  const float* xb = x + (size_t)b * NN * NN;
  for (int ir = wid; ir < RI; ir += 8) {
    const int grow = (rowBase + ir) & (NN - 1);
    const float* xr = xb + (size_t)grow * NN;
    float* dst = &xt[ir * XSTR];
    for (int jc = lane; jc < XSTR; jc += 32) {
      const int gcol = (colBase + jc) & (NN - 1);
      dst[jc] = xr[gcol];
    }
  }
  __syncthreads();

  // ---- stage 1: filter along columns j -> cA, cD tiles in LDS ----
  for (int o = tid; o < RI * CTILE; o += 256) {
    const int ir = o >> 6;         // CTILE == 64
    const int jc = o & 63;
    const float* xr = &xt[ir * XSTR + 2 * jc];
    float a = 0.f, d = 0.f;
#pragma unroll
    for (int t = 0; t < 8; ++t) {
      a += decLo[t] * xr[t + sLo];
      d += decHi[t] * xr[t + sHi];
    }
    cAt[ir * CSTR + jc] = a;
    cDt[ir * CSTR + jc] = d;
  }
  __syncthreads();

  // ---- stage 2: filter along rows i as banded matmul via WMMA f32 16x16x4 ----
  // quadrants: 0=AA (cA,lo) 1=AD (cA,hi) 2=DA (cD,lo) 3=DD (cD,hi)
  const int q   = wid >> 1;
  const int cs0 = (wid & 1) * 2;                 // two 16-col sub-tiles per wave
  const float* __restrict__ srcT = (q < 2) ? cAt : cDt;
  const float* __restrict__ dec  = (q & 1) ? decHi : decLo;
  const int s = (q & 1) ? sHi : sLo;

  // A-fragment lane mapping (16x4 f32): lane M = lane&15; lanes 16-31 hold K+2/K+3
  const int M  = lane & 15;
  const int kh = (lane >> 4) << 1;

  // Build banded weight fragments once: W[r][k] = dec[k - 2r - s] if in [0,8)
  v2f A[KCH];
#pragma unroll
  for (int c = 0; c < KCH; ++c) {
    const int k0 = 4 * c + kh;
    const int u0 = k0 - 2 * M - s;
    const int u1 = u0 + 1;
    A[c].x = (u0 >= 0 && u0 < 8) ? dec[u0] : 0.0f;
    A[c].y = (u1 >= 0 && u1 < 8) ? dec[u1] : 0.0f;
  }

  const int rowOutBase = ((q >= 2) ? HALF : 0) + r0 + (kh << 2); // + (lane>>4)*8
  float* outB = out + (size_t)b * NN * NN;

  for (int rep = 0; rep < 2; ++rep) {
    const int nc0 = (cs0 + rep) * 16;
    v8f acc = {0.f, 0.f, 0.f, 0.f, 0.f, 0.f, 0.f, 0.f};
#pragma unroll
    for (int c = 0; c < KCH; ++c) {
      const int k0 = 4 * c + kh;                 // B rows K / K+1 (upper lanes K+2/K+3)
      v2f Bf;
      Bf.x = srcT[k0 * CSTR + nc0 + M];
      Bf.y = srcT[(k0 + 1) * CSTR + nc0 + M];
      acc = __builtin_amdgcn_wmma_f32_16x16x4_f32(false, A[c], false, Bf,
                                                  (short)0, acc, false, false);
    }
    const int colOut = ((q & 1) ? HALF : 0) + c0 + nc0 + M;
    float* op = outB + (size_t)rowOutBase * NN + colOut;
#pragma unroll
    for (int v = 0; v < 8; ++v) {
      op[(size_t)v * NN] = acc[v];               // D: row = base + v, col = lane&15
    }
  }
}

extern "C" void kernel_launch(void* const* d_in, const int* in_sizes, int n_in,
                              void* d_out, int out_size, void* d_ws, size_t ws_size,
                              hipStream_t stream) {
  const float* x   = (const float*)d_in[0];
  const float* bmt = (const float*)d_in[1];
  const int*   m1  = (const int*)d_in[2];
  const int*   m2  = (const int*)d_in[3];
  float* out = (float*)d_out;

  const int B = in_sizes[0] / (NN * NN);   // 32
  dim3 grid(HALF / CTILE, HALF / RTILE, B);
  dim3 block(256);
  wav2d_kernel<<<grid, block, 0, stream>>>(x, bmt, m1, m2, out);

  (void)n_in; (void)out_size; (void)d_ws; (void)ws_size;
}